// Prompt_47871705481491
// MI455X (gfx1250) — compile-verified
//
#include <hip/hip_runtime.h>
#include <math.h>

typedef __attribute__((ext_vector_type(2))) float v2f;
typedef __attribute__((ext_vector_type(8))) float v8f;

#define Bsz  128
#define POOL 100
#define PPAD 112
#define LEN  25
#define Cdim 256
#define Ddim 256
#define Ktop 64

// ---------------- kernel 1: L2 normalize rows -------------------------------
// blocks 0..127  -> xn[row]   from cls_features
// blocks 128..239-> keyn[row-128] from prompt_key (rows >=100 zero-padded)
__global__ void norm_kernel(const float* __restrict__ cls,
                            const float* __restrict__ pkey,
                            float* __restrict__ xn,
                            float* __restrict__ keyn) {
    __shared__ float s[256];
    const int row = blockIdx.x;
    const int tid = threadIdx.x;

    const float* src;
    float* dst;
    bool pad = false;
    if (row < Bsz) {
        src = cls + (size_t)row * Ddim;
        dst = xn  + (size_t)row * Ddim;
    } else {
        int kr = row - Bsz;
        dst = keyn + (size_t)kr * Ddim;
        if (kr < POOL) src = pkey + (size_t)kr * Ddim;
        else { src = pkey; pad = true; }
    }
    float v = pad ? 0.0f : src[tid];
    s[tid] = v * v;
    __syncthreads();
    for (int st = 128; st > 0; st >>= 1) {
        if (tid < st) s[tid] += s[tid + st];
        __syncthreads();
    }
    float scale = rsqrtf(fmaxf(s[0], 1e-12f));
    dst[tid] = pad ? 0.0f : v * scale;
}

// ---------------- kernel 2: init scalar outputs -----------------------------
__global__ void init_kernel(float* __restrict__ rsim, float* __restrict__ pool) {
    int t = threadIdx.x;
    if (t == 0) rsim[0] = 0.0f;
    if (t < POOL) pool[t] = 0.0f;
}

// ---------------- kernel 3: similarity via V_WMMA_F32_16X16X4_F32 -----------
// grid (7, 8), 32 threads (one wave) per 16x16 tile of [128 x 112]
__global__ void sim_wmma_kernel(const float* __restrict__ xn,
                                const float* __restrict__ keyn,
                                float* __restrict__ sim) {
    const int lane = threadIdx.x;        // 0..31
    const int hf   = lane >> 4;          // 0 or 1 (half-wave)
    const int lrow = lane & 15;
    const int nBase = blockIdx.x * 16;   // 0..96
    const int mBase = blockIdx.y * 16;   // 0..112

    const int m = mBase + lrow;          // A row for this lane
    const int n = nBase + lrow;          // B col for this lane

    v8f acc = {};
    const float* arow = xn   + (size_t)m * Ddim;
    const float* brow = keyn + (size_t)n * Ddim;

    #pragma unroll 4
    for (int kk = 0; kk < Ddim; kk += 4) {
        // A 16x4 f32: lanes 0-15 -> K={kk,kk+1}; lanes 16-31 -> K={kk+2,kk+3}
        v2f a, b;
        a.x = arow[kk + 2 * hf + 0];
        a.y = arow[kk + 2 * hf + 1];
        // B 4x16 f32 (= keyn^T tile): same K striping, N across lanes
        b.x = brow[kk + 2 * hf + 0];
        b.y = brow[kk + 2 * hf + 1];
        acc = __builtin_amdgcn_wmma_f32_16x16x4_f32(
            false, a, false, b, (short)0, acc, false, false);
    }

    // C/D layout: VGPR r -> M = mBase + r + 8*hf, N = nBase + lrow
    const int col = nBase + lrow;
    if (col < POOL) {
        #pragma unroll
        for (int r = 0; r < 8; ++r) {
            int row = mBase + r + 8 * hf;
            sim[(size_t)row * POOL + col] = acc[r];
        }
    }
}

// ---------------- kernel 4: top-64 of 100 per row ---------------------------
// one block (128 threads) per batch row; repeated argmax with LDS tree reduce.
__global__ void topk_kernel(const float* __restrict__ sim,
                            int* __restrict__ idxm,
                            float* __restrict__ idxf,
                            float* __restrict__ pool,
                            float* __restrict__ rsim) {
    __shared__ float vals[128];
    __shared__ float rv[128];
    __shared__ int   ri[128];
    const int b   = blockIdx.x;
    const int tid = threadIdx.x;

    vals[tid] = (tid < POOL) ? sim[(size_t)b * POOL + tid] : -INFINITY;
    __syncthreads();

    float gsum = 0.0f;
    for (int kk = 0; kk < Ktop; ++kk) {
        rv[tid] = vals[tid];
        ri[tid] = tid;
        __syncthreads();
        for (int st = 64; st > 0; st >>= 1) {
            if (tid < st) {
                float v2 = rv[tid + st]; int i2 = ri[tid + st];
                float v1 = rv[tid];      int i1 = ri[tid];
                if (v2 > v1 || (v2 == v1 && i2 < i1)) { rv[tid] = v2; ri[tid] = i2; }
            }
            __syncthreads();
        }
        if (tid == 0) {
            int p = ri[0];
            gsum += rv[0];
            idxm[b * Ktop + kk] = p;
            idxf[b * Ktop + kk] = (float)p;
            vals[p] = -INFINITY;
            atomicAdd(&pool[p], 1.0f);
        }
        __syncthreads();
    }
    if (tid == 0) atomicAdd(rsim, gsum / (float)Bsz);
}

// ---------------- kernel 5: heavy bandwidth kernel (LDS-staged gather) ------
// One block per (b,k). Stage prompt[idx[b,k]] (25 KB) into LDS with coalesced
// reads (transposed to [c][l], stride-25 writes are bank-conflict-free), then
// stream out[b,c,k,l] = lds[c*25+l] + x[b,c,k,l].
__global__ void __launch_bounds__(256)
add_kernel(const float* __restrict__ x,
           const float* __restrict__ prompt,
           const int* __restrict__ idxm,
           float* __restrict__ out) {
    __shared__ float sp[LEN * Cdim];            // 6400 floats = 25.6 KB

    const int bk = blockIdx.x;                  // 0..8191
    const int b  = bk >> 6;
    const int k  = bk & 63;
    const int tid = threadIdx.x;

    const int p = idxm[bk];                     // uniform -> scalar load
    const float* prow = prompt + (size_t)p * (LEN * Cdim);

    // stage: coalesced global read (f contiguous), transposed LDS write
    #pragma unroll
    for (int it = 0; it < LEN; ++it) {          // 6400 / 256 = 25 iters
        int f = tid + it * 256;                 // f = l*256 + c
        float v = prow[f];
        int l = f >> 8;
        int c = f & 255;
        sp[c * LEN + l] = v;                    // stride 25 (odd) -> no conflicts
    }
    __syncthreads();

    // emit: out[b,c,k,l] = sp[c*25+l] + x[b,c,k,l]
    const size_t base = (size_t)b * (Cdim * Ktop * LEN) + (size_t)k * LEN;
    #pragma unroll
    for (int it = 0; it < LEN; ++it) {
        int e = tid + it * 256;                 // e = c*25 + l
        unsigned c = (unsigned)e / LEN;
        unsigned l = (unsigned)e - c * LEN;
        size_t g = base + (size_t)c * (Ktop * LEN) + l;
        out[g] = sp[e] + x[g];
    }
}

// ---------------- launch ----------------------------------------------------
extern "C" void kernel_launch(void* const* d_in, const int* in_sizes, int n_in,
                              void* d_out, int out_size, void* d_ws, size_t ws_size,
                              hipStream_t stream) {
    const float* x_embed = (const float*)d_in[0];   // [128,256,64,25]
    const float* cls     = (const float*)d_in[1];   // [128,256]
    const float* prompt  = (const float*)d_in[2];   // [100,25,256]
    const float* pkey    = (const float*)d_in[3];   // [100,256]

    float* out  = (float*)d_out;
    float* pe   = out;                              // 52,428,800
    float* rsim = out + 52428800;                   // 1
    float* sim  = out + 52428801;                   // 12,800
    float* idxf = out + 52441601;                   // 8,192
    float* pool = out + 52449793;                   // 100

    float* wsf  = (float*)d_ws;
    float* xn   = wsf;                              // 128*256 = 32768 f32
    float* keyn = wsf + 32768;                      // 112*256 = 28672 f32
    int*   idxm = (int*)(wsf + 61440);              // 128*64 ints

    norm_kernel<<<Bsz + PPAD, 256, 0, stream>>>(cls, pkey, xn, keyn);
    init_kernel<<<1, 128, 0, stream>>>(rsim, pool);
    sim_wmma_kernel<<<dim3(PPAD / 16, Bsz / 16), 32, 0, stream>>>(xn, keyn, sim);
    topk_kernel<<<Bsz, 128, 0, stream>>>(sim, idxm, idxf, pool, rsim);

    add_kernel<<<Bsz * Ktop, 256, 0, stream>>>(x_embed, prompt, idxm, pe);
}